// SpatialTransformer_80461917324002
// MI455X (gfx1250) — compile-verified
//
#include <hip/hip_runtime.h>

// Spatial transformer (affine grid sample, bilinear) for MI455X / gfx1250.
// B=16, H=512, W=512, C=16, all f32.
// Memory-bound: ~512MB min traffic @ 23.3 TB/s => ~22us floor. No WMMA-worthy
// matrix op exists (theta x grid is 2 FMAs/pixel); kernel is pure data movement.

typedef float v4f __attribute__((ext_vector_type(4)));

#define B_ 16
#define H_ 512
#define W_ 512
#define C_ 16
#define HW_ (H_ * W_)

// 256 threads = 8 waves; 4 lanes per pixel (one float4 channel-group each);
// 64 pixels per iteration, 4 iterations => 256 pixels per block.
__global__ __launch_bounds__(256) void st_bilinear_kernel(
    const float* __restrict__ images,
    const float* __restrict__ theta,
    float* __restrict__ out) {
  const int b = blockIdx.y;

  // Uniform per block -> scalar loads.
  const float* __restrict__ th = theta + b * 6;
  const float t00 = th[0], t01 = th[1], t02 = th[2];
  const float t10 = th[3], t11 = th[4], t12 = th[5];

  const int cg   = threadIdx.x & 3;   // channel group: 4 floats each
  const int slot = threadIdx.x >> 2;  // pixel slot within iteration: 0..63

  const v4f* __restrict__ img4 = (const v4f*)images + (size_t)b * HW_ * 4;
  v4f* __restrict__ out4       = (v4f*)out          + (size_t)b * HW_ * 4;

  int pix = blockIdx.x * 256 + slot;

#pragma unroll
  for (int it = 0; it < 4; ++it, pix += 64) {
    const int px = pix & (W_ - 1);
    const int py = pix >> 9;  // W_ == 512

    // t_g = theta . [px, py, 1]
    const float xt = t00 * (float)px + t01 * (float)py + t02;
    const float yt = t10 * (float)px + t11 * (float)py + t12;

    // map to pixel space
    const float x = 0.5f * ((xt + 1.0f) * (float)W_ - 1.0f);
    const float y = 0.5f * ((yt + 1.0f) * (float)H_ - 1.0f);

    // corner indices: floor, +1, then clip (reference order)
    const float fx = floorf(x);
    const float fy = floorf(y);
    int x0 = (int)fx, y0 = (int)fy;
    int x1 = x0 + 1,  y1 = y0 + 1;
    x0 = min(max(x0, 0), W_ - 1);
    x1 = min(max(x1, 0), W_ - 1);
    y0 = min(max(y0, 0), H_ - 1);
    y1 = min(max(y1, 0), H_ - 1);

    // weights from CLIPPED corner coords (matches reference)
    const float x0f = (float)x0, x1f = (float)x1;
    const float y0f = (float)y0, y1f = (float)y1;
    const float wa = (x1f - x) * (y1f - y);  // p00 = (y0, x0)
    const float wb = (x1f - x) * (y - y0f);  // p01 = (y1, x0)
    const float wc = (x - x0f) * (y1f - y);  // p10 = (y0, x1)
    const float wd = (x - x0f) * (y - y0f);  // p11 = (y1, x1)

    const int r0 = y0 * W_;
    const int r1 = y1 * W_;
    const v4f* __restrict__ p00 = img4 + (size_t)(r0 + x0) * 4 + cg;
    const v4f* __restrict__ p01 = img4 + (size_t)(r1 + x0) * 4 + cg;
    const v4f* __restrict__ p10 = img4 + (size_t)(r0 + x1) * 4 + cg;
    const v4f* __restrict__ p11 = img4 + (size_t)(r1 + x1) * 4 + cg;

    if (it == 0) {
      // gfx1250 prefetch path (global_prefetch_b8); pulls first corner lines
      // into cache while weight math for later iterations is in flight.
      __builtin_prefetch(p00, 0, 3);
      __builtin_prefetch(p01, 0, 3);
      __builtin_prefetch(p10, 0, 3);
      __builtin_prefetch(p11, 0, 3);
    }

    const v4f a = *p00;
    const v4f c01 = *p01;
    const v4f c10 = *p10;
    const v4f d = *p11;

    const v4f r = wa * a + wb * c01 + wc * c10 + wd * d;

    // Output is write-once streaming: non-temporal so the 256MB result does not
    // evict the gather working set (16MB/image) from the 192MB L2.
    __builtin_nontemporal_store(r, out4 + (size_t)pix * 4 + cg);
  }
}

extern "C" void kernel_launch(void* const* d_in, const int* in_sizes, int n_in,
                              void* d_out, int out_size, void* d_ws, size_t ws_size,
                              hipStream_t stream) {
  const float* images = (const float*)d_in[0];  // (B,H,W,C) f32
  const float* theta  = (const float*)d_in[1];  // (B,2,3)   f32
  float* out = (float*)d_out;                   // (B*H*W, C) f32

  (void)in_sizes; (void)n_in; (void)out_size; (void)d_ws; (void)ws_size;

  // 256 pixels per block; HW_/256 = 1024 blocks in x, batch in y.
  dim3 grid(HW_ / 256, B_, 1);
  dim3 block(256, 1, 1);
  st_bilinear_kernel<<<grid, block, 0, stream>>>(images, theta, out);
}